// Model_11879879542469
// MI455X (gfx1250) — compile-verified
//
#include <hip/hip_runtime.h>
#include <math.h>

// ---------------------------------------------------------------------------
// Reference pipeline:
//   y = x^T x            (4M x 10) -> (10 x 10)      <- ~160 MB read, dominant
//   v, w = eigh(y)       eigenvectors, ascending eigenvalues
//   h = relu(w @ W1^T + b1); o = h @ W2^T + b2; out = 0.5*(sigmoid(o)+1)
//
// Phase 1: memory-bound Gram reduction using V_WMMA_F32_16X16X4_F32.
//   For Gram = X^T X, the A-matrix (16x4, M=dim,K=row) and B-matrix
//   (4x16, K=row,N=dim) per-lane layouts coincide:
//     lane L, VGPR v  ->  x[row = 4*c + v + 2*(L>=16), col = L & 15]
//   so one 2-VGPR load feeds both WMMA operands. Lanes with col>=10 load a
//   clamped in-bounds address and are zeroed with a select (no exec masking).
//   Each wave owns a contiguous range of full 4-row chunks -> branch-free
//   inner loop. N%4 leftover rows are folded in scalar-style in phase 2.
// Phase 2: deterministic partial-sum reduce + f64 Jacobi eigh + tiny MLP.
// ---------------------------------------------------------------------------

typedef __attribute__((ext_vector_type(2))) float v2f;
typedef __attribute__((ext_vector_type(8))) float v8f;

#define WAVES_PER_WG 8
#define THREADS (WAVES_PER_WG * 32)
#define UNROLL 8

__global__ __launch_bounds__(THREADS) void gram_phase1(
    const float* __restrict__ x, int N, float* __restrict__ ws, int G) {
  const int lane = threadIdx.x & 31;
  const int wave = threadIdx.x >> 5;
  const int col  = lane & 15;            // dimension index (valid < 10)
  const int half = lane >> 4;            // 0 -> K rows 0,1 ; 1 -> K rows 2,3
  const bool colOK = (col < 10);
  const int colc = colOK ? col : 0;      // clamped, always in-bounds

  const int totalWaves  = G * WAVES_PER_WG;
  const int mainChunks  = N >> 2;        // full 4-row chunks only
  const int cpw         = (mainChunks + totalWaves - 1) / totalWaves;
  const int waveGlobal  = blockIdx.x * WAVES_PER_WG + wave;

  int beg = waveGlobal * cpw;
  int end = beg + cpw;
  if (beg > mainChunks) beg = mainChunks;
  if (end > mainChunks) end = mainChunks;

  // Per-lane element offset inside a chunk (chunk = 4 rows = 40 floats).
  const int off = half * 20 + colc;      // row (half*2) * 10 + col

  v8f acc = {0.f, 0.f, 0.f, 0.f, 0.f, 0.f, 0.f, 0.f};

  int c = beg;
  // ---- unrolled main loop: branch-free, unconditional loads ----
  for (; c + UNROLL <= end; c += UNROLL) {
    const float* __restrict__ pc = x + (size_t)c * 40;
    v2f a[UNROLL];
#pragma unroll
    for (int u = 0; u < UNROLL; ++u) {
      float f0 = pc[u * 40 + off];        // row 4c+4u + half*2
      float f1 = pc[u * 40 + off + 10];   // row 4c+4u + half*2 + 1
      a[u][0] = colOK ? f0 : 0.f;
      a[u][1] = colOK ? f1 : 0.f;
    }
#pragma unroll
    for (int u = 0; u < UNROLL; ++u) {
      // D = A * B + C ; A == B register image (see header comment)
      acc = __builtin_amdgcn_wmma_f32_16x16x4_f32(
          false, a[u], false, a[u], (short)0, acc, false, false);
    }
  }
  // ---- tail chunks (< UNROLL), still branch-free per load ----
  for (; c < end; ++c) {
    const float* __restrict__ pc = x + (size_t)c * 40;
    v2f a;
    float f0 = pc[off];
    float f1 = pc[off + 10];
    a[0] = colOK ? f0 : 0.f;
    a[1] = colOK ? f1 : 0.f;
    acc = __builtin_amdgcn_wmma_f32_16x16x4_f32(
        false, a, false, a, (short)0, acc, false, false);
  }

  // Reduce the 8 waves' accumulators via LDS, extract the 10x10 block.
  // C/D layout: VGPR v, lanes 0-15 -> (M=v, N=lane); lanes 16-31 -> (M=v+8).
  __shared__ float lds[WAVES_PER_WG * 256];
#pragma unroll
  for (int v = 0; v < 8; ++v)
    lds[wave * 256 + v * 32 + lane] = acc[v];
  __syncthreads();

  if (threadIdx.x < 100) {
    const int m = threadIdx.x / 10, n = threadIdx.x % 10;
    const int v  = m & 7;
    const int ln = ((m >> 3) << 4) + n;
    float s = 0.f;
#pragma unroll
    for (int w = 0; w < WAVES_PER_WG; ++w)
      s += lds[w * 256 + v * 32 + ln];
    ws[blockIdx.x * 128 + m * 10 + n] = s;   // 128-float slot per workgroup
  }
}

__global__ __launch_bounds__(128) void gram_eigh_mlp(
    const float* __restrict__ ws, int G,
    const float* __restrict__ x, int N,
    const float* __restrict__ W1, const float* __restrict__ b1,
    const float* __restrict__ W2, const float* __restrict__ b2,
    float* __restrict__ out) {
  __shared__ float gram[100];
  const int t = threadIdx.x;
  if (t < 100) {
    float s = 0.f;
    for (int g = 0; g < G; ++g) s += ws[g * 128 + t];  // fixed order: deterministic
    // leftover rows not covered by 4-row chunks (N % 4 rows)
    const int m = t / 10, n = t % 10;
    for (int r = (N >> 2) << 2; r < N; ++r)
      s += x[r * 10 + m] * x[r * 10 + n];
    gram[t] = s;
  }
  __syncthreads();
  if (t != 0) return;

  // --- Cyclic Jacobi eigensolver on the 10x10 symmetric Gram (f64) ---
  double A[10][10], V[10][10];
  for (int i = 0; i < 10; ++i)
    for (int j = 0; j < 10; ++j) {
      A[i][j] = (double)gram[i * 10 + j];
      V[i][j] = (i == j) ? 1.0 : 0.0;
    }
  for (int sweep = 0; sweep < 12; ++sweep) {
    for (int p = 0; p < 9; ++p) {
      for (int q = p + 1; q < 10; ++q) {
        const double apq = A[p][q];
        if (fabs(apq) <= 1e-30 * (fabs(A[p][p]) + fabs(A[q][q]) + 1e-300))
          continue;
        const double tau = (A[q][q] - A[p][p]) / (2.0 * apq);
        const double tt  = (tau >= 0.0) ? 1.0 / (tau + sqrt(1.0 + tau * tau))
                                        : 1.0 / (tau - sqrt(1.0 + tau * tau));
        const double cc = 1.0 / sqrt(1.0 + tt * tt);
        const double ss = tt * cc;
        for (int k = 0; k < 10; ++k) {  // column rotation
          const double a1 = A[k][p], a2 = A[k][q];
          A[k][p] = cc * a1 - ss * a2;
          A[k][q] = ss * a1 + cc * a2;
        }
        for (int k = 0; k < 10; ++k) {  // row rotation
          const double a1 = A[p][k], a2 = A[q][k];
          A[p][k] = cc * a1 - ss * a2;
          A[q][k] = ss * a1 + cc * a2;
        }
        for (int k = 0; k < 10; ++k) {  // accumulate eigenvectors (columns)
          const double v1 = V[k][p], v2 = V[k][q];
          V[k][p] = cc * v1 - ss * v2;
          V[k][q] = ss * v1 + cc * v2;
        }
      }
    }
  }

  // Sort eigenpairs ascending (matches jnp.linalg.eigh ordering).
  double ev[10]; int idx[10];
  for (int i = 0; i < 10; ++i) { ev[i] = A[i][i]; idx[i] = i; }
  for (int i = 0; i < 9; ++i) {
    int mi = i;
    for (int j = i + 1; j < 10; ++j)
      if (ev[idx[j]] < ev[idx[mi]]) mi = j;
    const int tmp = idx[i]; idx[i] = idx[mi]; idx[mi] = tmp;
  }

  // Eigenvector sign is arbitrary in eigh; normalize: largest-|entry| positive.
  float w[10][10];
  for (int d = 0; d < 10; ++d) {
    const int cI = idx[d];
    double mx = 0.0; int mi = 0;
    for (int i = 0; i < 10; ++i)
      if (fabs(V[i][cI]) > mx) { mx = fabs(V[i][cI]); mi = i; }
    const double sgn = (V[mi][cI] >= 0.0) ? 1.0 : -1.0;
    for (int i = 0; i < 10; ++i)
      w[i][d] = (float)(V[i][cI] * sgn);
  }

  // h = relu(w @ W1^T + b1) -> (10,16); o = h @ W2^T + b2 -> (10,1)
  for (int i = 0; i < 10; ++i) {
    float o = b2[0];
    for (int j = 0; j < 16; ++j) {
      float h = b1[j];
      for (int d = 0; d < 10; ++d) h += w[i][d] * W1[j * 10 + d];
      h = (h > 0.f) ? h : 0.f;
      o += h * W2[j];
    }
    out[i] = 0.5f * (1.f / (1.f + expf(-o)) + 1.f);
  }
}

extern "C" void kernel_launch(void* const* d_in, const int* in_sizes, int n_in,
                              void* d_out, int out_size, void* d_ws, size_t ws_size,
                              hipStream_t stream) {
  (void)n_in; (void)out_size;
  const float* x  = (const float*)d_in[0];
  const float* W1 = (const float*)d_in[1];
  const float* b1 = (const float*)d_in[2];
  const float* W2 = (const float*)d_in[3];
  const float* b2 = (const float*)d_in[4];
  float* out = (float*)d_out;
  float* ws  = (float*)d_ws;

  const int N = in_sizes[0] / 10;

  // 2048 WGs x 256 threads targets HBM saturation; clamp to workspace size
  // (deterministic given the same ws_size).
  int G = 2048;
  const size_t slot = 128 * sizeof(float);
  if ((size_t)G * slot > ws_size) G = (int)(ws_size / slot);
  if (G < 1) G = 1;

  gram_phase1<<<G, THREADS, 0, stream>>>(x, N, ws, G);
  gram_eigh_mlp<<<1, 128, 0, stream>>>(ws, G, x, N, W1, b1, W2, b2, out);
}